// IoU3DLoss_67319317397956
// MI455X (gfx1250) — compile-verified
//
#include <hip/hip_runtime.h>
#include <cstdint>

#define BS 128
#define EPSF 1e-8f

typedef __attribute__((ext_vector_type(4))) unsigned int u32x4;
typedef __attribute__((ext_vector_type(4))) int i32x4;
typedef __attribute__((ext_vector_type(8))) int i32x8;

#if defined(__gfx1250__) && __has_builtin(__builtin_amdgcn_tensor_load_to_lds)
#define HAVE_TDM 1
#endif

__device__ __forceinline__ unsigned umin_(unsigned a, unsigned b) { return a < b ? a : b; }

#if defined(HAVE_TDM)
// 1-D TDM tile load: tile_elems f32/i32 elements from gsrc -> LDS at ldst.
// tensor_elems clamps the valid extent (OOB reads return zero per ISA §8.2).
__device__ __forceinline__ void tdm_load_1d(const void* gsrc, void* ldst,
                                            unsigned tile_elems, unsigned tensor_elems) {
  unsigned long long ga = (unsigned long long)(uintptr_t)gsrc;
  unsigned lds = (unsigned)(uintptr_t)ldst;  // low 32 bits of generic shared ptr == LDS byte offset
  u32x4 g0;
  g0[0] = 1u;                                        // count=1 (valid user descriptor)
  g0[1] = lds;                                       // lds_addr
  g0[2] = (unsigned)(ga & 0xffffffffull);            // global_addr[31:0]
  g0[3] = (unsigned)((ga >> 32) & 0x01ffffffull) | 0x80000000u;  // addr[56:32] | type=2
  i32x8 g1;
  g1[0] = 0x00020000;                                // data_size=2 (4 bytes)
  g1[1] = (int)((tensor_elems & 0xffffu) << 16);     // tensor_dim0[15:0]
  g1[2] = (int)(((tensor_elems >> 16) & 0xffffu) | (1u << 16)); // tensor_dim0[31:16], tensor_dim1=1
  g1[3] = (int)((tile_elems & 0xffffu) << 16);       // tile_dim0
  g1[4] = 0;                                         // tile_dim1=0, tile_dim2=0 (1-D tile)
  g1[5] = (int)tensor_elems;                         // tensor_dim0_stride (unused for 1-D)
  g1[6] = 0;
  g1[7] = 0;
  i32x4 gz = {0, 0, 0, 0};
#if __clang_major__ >= 23
  i32x8 gz8 = {0, 0, 0, 0, 0, 0, 0, 0};
  __builtin_amdgcn_tensor_load_to_lds(g0, g1, gz, gz, gz8, 0);
#else
  __builtin_amdgcn_tensor_load_to_lds(g0, g1, gz, gz, 0);
#endif
}
#endif

__global__ void __launch_bounds__(BS) iou3d_main(
    const float* __restrict__ pred_iou, const float* __restrict__ pred_boxes,
    const float* __restrict__ gt_boxes, const int* __restrict__ mask,
    float* __restrict__ partial, int n) {
  __shared__ float sPred[BS * 7];
  __shared__ float sGt[BS * 8];
  __shared__ float sIou[BS];
  __shared__ int sMask[BS];
  __shared__ float sVx[24 * BS];   // column-per-lane: index k*BS+tid (bank-conflict-free)
  __shared__ float sVy[24 * BS];
  __shared__ float sAng[24 * BS];
  __shared__ float sRed[8];

  const int tid = threadIdx.x;
  const int block0 = blockIdx.x * BS;
  const int rem = n - block0;

#if defined(HAVE_TDM)
  {
    const int w = tid >> 5;  // one TDM descriptor per wave (4 waves, 4 arrays)
    const unsigned remu = (unsigned)(rem < 0 ? 0 : rem);
    if (w == 0) {
      tdm_load_1d(pred_boxes + (size_t)block0 * 7, sPred, BS * 7u, umin_(BS * 7u, remu * 7u));
    } else if (w == 1) {
      tdm_load_1d(gt_boxes + (size_t)block0 * 8, sGt, BS * 8u, umin_(BS * 8u, remu * 8u));
    } else if (w == 2) {
      tdm_load_1d(pred_iou + block0, sIou, (unsigned)BS, umin_((unsigned)BS, remu));
    } else {
      tdm_load_1d(mask + block0, sMask, (unsigned)BS, umin_((unsigned)BS, remu));
    }
    __builtin_amdgcn_s_wait_tensorcnt(0);  // each wave waits on its own TENSORcnt
  }
#else
  for (int e = tid; e < BS * 7; e += BS) {
    long long ge = (long long)block0 * 7 + e;
    sPred[e] = (ge < (long long)n * 7) ? pred_boxes[ge] : 0.f;
  }
  for (int e = tid; e < BS * 8; e += BS) {
    long long ge = (long long)block0 * 8 + e;
    sGt[e] = (ge < (long long)n * 8) ? gt_boxes[ge] : 0.f;
  }
  {
    int g = block0 + tid;
    sIou[tid] = (g < n) ? pred_iou[g] : 0.f;
    sMask[tid] = (g < n) ? mask[g] : 0;
  }
#endif
  __syncthreads();

  float contrib = 0.f, mval = 0.f;
  const int i = block0 + tid;
  if (i < n) {
    const float px = sPred[tid * 7 + 0], py = sPred[tid * 7 + 1], pz = sPred[tid * 7 + 2];
    const float pw = sPred[tid * 7 + 3], ph = sPred[tid * 7 + 4], pl = sPred[tid * 7 + 5];
    const float pa = sPred[tid * 7 + 6];
    const float gx = sGt[tid * 8 + 0], gy = sGt[tid * 8 + 1], gzc = sGt[tid * 8 + 2];
    const float gw = sGt[tid * 8 + 3], gh = sGt[tid * 8 + 4], gl = sGt[tid * 8 + 5];
    const float gsn = sGt[tid * 8 + 6], gcs = sGt[tid * 8 + 7];
    const float ga = atan2f(gsn, gcs);

    float c1x[4], c1y[4], c2x[4], c2y[4];
    {
      const float dxs[4] = {0.5f, -0.5f, -0.5f, 0.5f};
      const float dys[4] = {0.5f, 0.5f, -0.5f, -0.5f};
      const float ca = cosf(pa), sa = sinf(pa);
      const float cb = cosf(ga), sb = sinf(ga);
#pragma unroll
      for (int k = 0; k < 4; k++) {
        float cx = dxs[k] * pw, cy = dys[k] * ph;
        c1x[k] = cx * ca - cy * sa + px;
        c1y[k] = cx * sa + cy * ca + py;
        float ex = dxs[k] * gw, ey = dys[k] * gh;
        c2x[k] = ex * cb - ey * sb + gx;
        c2y[k] = ex * sb + ey * cb + gy;
      }
    }

    auto inbox = [](float mx, float my, const float* qx, const float* qy) -> bool {
      float ax = qx[0], ay = qy[0];
      float abx = qx[1] - ax, aby = qy[1] - ay;
      float adx = qx[3] - ax, ady = qy[3] - ay;
      float amx = mx - ax, amy = my - ay;
      float pab = (abx * amx + aby * amy) / (abx * abx + aby * aby);
      float pad = (adx * amx + ady * amy) / (adx * adx + ady * ady);
      return (pab > -1e-6f) && (pab < 1.000001f) && (pad > -1e-6f) && (pad < 1.000001f);
    };

    unsigned vbits = 0u;
    float sumx = 0.f, sumy = 0.f;
    int cnt = 0;

#pragma unroll
    for (int k = 0; k < 4; k++) {  // box1 corners
      sVx[k * BS + tid] = c1x[k];
      sVy[k * BS + tid] = c1y[k];
      if (inbox(c1x[k], c1y[k], c2x, c2y)) { vbits |= 1u << k; sumx += c1x[k]; sumy += c1y[k]; cnt++; }
    }
#pragma unroll
    for (int k = 0; k < 4; k++) {  // box2 corners
      sVx[(4 + k) * BS + tid] = c2x[k];
      sVy[(4 + k) * BS + tid] = c2y[k];
      if (inbox(c2x[k], c2y[k], c1x, c1y)) { vbits |= 1u << (4 + k); sumx += c2x[k]; sumy += c2y[k]; cnt++; }
    }
#pragma unroll
    for (int a = 0; a < 4; a++) {  // 16 edge-pair intersections
      float x1 = c1x[a], y1 = c1y[a], x2 = c1x[(a + 1) & 3], y2 = c1y[(a + 1) & 3];
#pragma unroll
      for (int b = 0; b < 4; b++) {
        float x3 = c2x[b], y3 = c2y[b], x4 = c2x[(b + 1) & 3], y4 = c2y[(b + 1) & 3];
        float num = (x2 - x1) * (y3 - y4) - (y2 - y1) * (x3 - x4);
        float dent = (x1 - x3) * (y3 - y4) - (y1 - y3) * (x3 - x4);
        float denu = (x1 - x3) * (y1 - y2) - (y1 - y3) * (x1 - x2);
        float t = dent / (num + EPSF);
        float u = -denu / (num + EPSF);
        bool v = (t > 0.f) && (t < 1.f) && (u > 0.f) && (u < 1.f) && (num != 0.f);
        float ix = x1 + t * (x2 - x1);
        float iy = y1 + t * (y2 - y1);
        int slot = 8 + a * 4 + b;
        sVx[slot * BS + tid] = ix;
        sVy[slot * BS + tid] = iy;
        if (v) { vbits |= 1u << slot; sumx += ix; sumy += iy; cnt++; }
      }
    }

    const int denom = (cnt < 1) ? 1 : cnt;
    const float meanx = sumx / (float)denom, meany = sumy / (float)denom;
    const float INF = __builtin_inff();

#pragma unroll 1
    for (int k = 0; k < 24; k++) {  // center + angles (atan2 only for valid)
      float vx = sVx[k * BS + tid] - meanx;
      float vy = sVy[k * BS + tid] - meany;
      sVx[k * BS + tid] = vx;
      sVy[k * BS + tid] = vy;
      float a2 = INF;
      if ((vbits >> k) & 1u) a2 = atan2f(vy, vx);
      sAng[k * BS + tid] = a2;
    }

    // global minimum key (== first vertex of the stable angle-sort)
    float gmna = INF;
    int gmni = 0;
#pragma unroll 1
    for (int k = 0; k < 24; k++) {
      if ((vbits >> k) & 1u) {
        float a2 = sAng[k * BS + tid];
        if (a2 < gmna) { gmna = a2; gmni = k; }
      }
    }

    // Shoelace over the angle-sorted chain via lexicographic-(angle,idx) successor.
    // Equal-angle (parallel) vertex pairs have zero cross product, so this equals
    // the reference's stable-argsort + duplicated-tail formulation exactly.
    float csum = 0.f;
#pragma unroll 1
    for (int k = 0; k < 24; k++) {
      if (!((vbits >> k) & 1u)) continue;
      float ak = sAng[k * BS + tid];
      float ba = INF;
      int bi = -1;
#pragma unroll 1
      for (int j = 0; j < 24; j++) {
        if (!((vbits >> j) & 1u)) continue;
        float aj = sAng[j * BS + tid];
        bool gtk = (aj > ak) || ((aj == ak) && (j > k));
        bool ltb = (bi < 0) ? true : ((aj < ba) || ((aj == ba) && (j < bi)));
        if (gtk && ltb) { ba = aj; bi = j; }
      }
      if (bi < 0) bi = gmni;  // wraparound closes the polygon
      float kx = sVx[k * BS + tid], ky = sVy[k * BS + tid];
      float bx = sVx[bi * BS + tid], by = sVy[bi * BS + tid];
      csum += kx * by - ky * bx;
    }

    float inter2d = fabsf(csum) * 0.5f;
    float zmax = fminf(pz + pl * 0.5f, gzc + gl * 0.5f);
    float zmin = fmaxf(pz - pl * 0.5f, gzc - gl * 0.5f);
    float inter3d = inter2d * fmaxf(zmax - zmin, 0.f);
    float v1 = pw * ph * pl, v2 = gw * gh * gl;
    float iou3 = inter3d / (v1 + v2 - inter3d);
    float target = iou3 * 2.f - 1.f;
    mval = (float)sMask[tid];
    contrib = fabsf(sIou[tid] - target) * mval;
  }

  // wave32 shuffle reduction, then 4 partials -> block partial (deterministic)
  float c = contrib, mv = mval;
#pragma unroll
  for (int off = 16; off > 0; off >>= 1) {
    c += __shfl_down(c, off, 32);
    mv += __shfl_down(mv, off, 32);
  }
  const int wave = tid >> 5, lane = tid & 31;
  if (lane == 0) { sRed[wave] = c; sRed[4 + wave] = mv; }
  __syncthreads();
  if (tid == 0) {
    float tc = 0.f, tm = 0.f;
#pragma unroll
    for (int w = 0; w < BS / 32; w++) { tc += sRed[w]; tm += sRed[4 + w]; }
    partial[2 * blockIdx.x] = tc;
    partial[2 * blockIdx.x + 1] = tm;
  }
}

__global__ void __launch_bounds__(256) iou3d_finalize(const float* __restrict__ partial,
                                                      int nblocks, float* __restrict__ out) {
  __shared__ float s0[256], s1[256];
  float a = 0.f, b = 0.f;
  for (int i = threadIdx.x; i < nblocks; i += 256) {
    a += partial[2 * i];
    b += partial[2 * i + 1];
  }
  s0[threadIdx.x] = a;
  s1[threadIdx.x] = b;
  __syncthreads();
  for (int off = 128; off > 0; off >>= 1) {
    if ((int)threadIdx.x < off) {
      s0[threadIdx.x] += s0[threadIdx.x + off];
      s1[threadIdx.x] += s1[threadIdx.x + off];
    }
    __syncthreads();
  }
  if (threadIdx.x == 0) out[0] = s0[0] / fmaxf(s1[0], 1e-4f);  // * LOSS_WEIGHT(=1.0)
}

extern "C" void kernel_launch(void* const* d_in, const int* in_sizes, int n_in,
                              void* d_out, int out_size, void* d_ws, size_t ws_size,
                              hipStream_t stream) {
  const float* pred_iou = (const float*)d_in[0];
  const float* pred_boxes = (const float*)d_in[1];
  const float* gt_boxes = (const float*)d_in[2];
  const int* mask = (const int*)d_in[3];
  const int n = in_sizes[0];
  const int nblocks = (n + BS - 1) / BS;
  float* partial = (float*)d_ws;  // nblocks*2 floats
  iou3d_main<<<nblocks, BS, 0, stream>>>(pred_iou, pred_boxes, gt_boxes, mask, partial, n);
  iou3d_finalize<<<1, 256, 0, stream>>>(partial, nblocks, (float*)d_out);
}